// PatchMemoryBank_34102040330522
// MI455X (gfx1250) — compile-verified
//
#include <hip/hip_runtime.h>
#include <math.h>

// Problem constants (fixed by the reference).
#define NPTS   98304      // B*H*W
#define CDIM   256
#define HWSZ   1024       // H*W
#define PLANE  262144     // C*H*W
#define NSAMP  491
#define NITER  490        // NSAMP - 1 selections after index 0
#define PTS_PER_BLOCK 256 // 32 points per wave, 8 waves
#define DIST_BLOCKS (NPTS / PTS_PER_BLOCK)   // 384

typedef __attribute__((ext_vector_type(2))) float v2f;
typedef __attribute__((ext_vector_type(8))) float v8f;

__device__ __forceinline__ void combineMax(float& v, int& i, float ov, int oi) {
  // max value; ties -> lower index (matches jnp.argmax first-max semantics)
  if (ov > v || (ov == v && oi < i)) { v = ov; i = oi; }
}

__device__ __forceinline__ void waveReduceMax(float& v, int& i) {
#pragma unroll
  for (int off = 16; off >= 1; off >>= 1) {
    float ov = __shfl_xor(v, off, 32);
    int   oi = __shfl_xor(i, off, 32);
    combineMax(v, i, ov, oi);
  }
}

// Extract diag element for this lane from a 16x16 f32 accumulator:
// d2(i) lives in component i @ lane i (i<8) and component i-8 @ lane i+16 (i>=8).
__device__ __forceinline__ float diagExtract(const v8f acc, int lane) {
  const float a0 = acc[0], a1 = acc[1], a2 = acc[2], a3 = acc[3];
  const float a4 = acc[4], a5 = acc[5], a6 = acc[6], a7 = acc[7];
  const int r = lane & 7;
  float t0 = (r & 1) ? a1 : a0;
  float t1 = (r & 1) ? a3 : a2;
  float t2 = (r & 1) ? a5 : a4;
  float t3 = (r & 1) ? a7 : a6;
  float u0 = (r & 2) ? t1 : t0;
  float u1 = (r & 2) ? t3 : t2;
  float compVal = (r & 4) ? u1 : u0;
  float shifted = __shfl(compVal, lane + 16, 32);
  return (lane & 8) ? shifted : compVal;
}

__global__ void __launch_bounds__(256) coreset_init(float* __restrict__ min_d,
                                                    int* __restrict__ indices,
                                                    int* __restrict__ counter) {
  int i = blockIdx.x * 256 + threadIdx.x;
  if (i < NPTS) min_d[i] = __builtin_huge_valf();
  if (i == 0) { indices[0] = 0; *counter = 0; }
}

// One greedy round: distances from center indices[iter] to all points via
// WMMA diag(D * D^T) (two interleaved 16-point chains per wave),
// min_d update, global argmax -> indices[iter+1].
__global__ void __launch_bounds__(256) coreset_step(
    const float* __restrict__ F, float* __restrict__ min_d,
    float* __restrict__ pval, int* __restrict__ pidx,
    int* __restrict__ indices, int* __restrict__ counter, int iter) {
  __shared__ float qsh[CDIM];
  __shared__ float rv[8];
  __shared__ int   ri[8];
  __shared__ int   lastFlag;

  const int tid  = threadIdx.x;
  const int lane = tid & 31;
  const int wv   = tid >> 5;

  // Stage the query vector (strided gather along c-planes; L2-resident).
  const int qn = indices[iter];
  const int qb = qn >> 10, qp = qn & (HWSZ - 1);
  qsh[tid] = F[(size_t)qb * PLANE + (size_t)tid * HWSZ + qp];
  __syncthreads();

  // 32 points per wave (two 16-row WMMA chains). p0 is a multiple of 32 and
  // HWSZ % 32 == 0, so all 32 points share the same batch b.
  const int p0   = blockIdx.x * PTS_PER_BLOCK + wv * 32;
  const int b    = p0 >> 10;
  const int m    = lane & 15;     // matrix row M for lanes 0-15 (mirrored 16-31)
  const int hf   = lane >> 4;     // lane half selects K pair {0,1} vs {2,3}
  const int cofs = 2 * hf;
  // lane loads float2 -> points (p0+2m, p0+2m+1): chain A = even, chain B = odd
  const float* base = F + (size_t)b * PLANE + ((p0 & (HWSZ - 1)) + 2 * m);

  v8f accA = {}, accB = {};
  // software pipeline: preload chunk 0
  v2f x0 = *reinterpret_cast<const v2f*>(base + (size_t)(0 + cofs) * HWSZ);
  v2f x1 = *reinterpret_cast<const v2f*>(base + (size_t)(1 + cofs) * HWSZ);
  v2f qq = *reinterpret_cast<const v2f*>(qsh + cofs);

  // 63 pipelined chunks (c0 = 0..248); loads reach plane 255 max (in bounds).
#pragma unroll 3
  for (int c0 = 0; c0 < CDIM - 4; c0 += 4) {
    v2f nx0 = *reinterpret_cast<const v2f*>(base + (size_t)(c0 + 4 + cofs) * HWSZ);
    v2f nx1 = *reinterpret_cast<const v2f*>(base + (size_t)(c0 + 5 + cofs) * HWSZ);
    v2f nqq = *reinterpret_cast<const v2f*>(qsh + c0 + 4 + cofs);

    v2f Aa, Ab;
    Aa.x = x0.x - qq.x;  Aa.y = x1.x - qq.y;   // even points
    Ab.x = x0.y - qq.x;  Ab.y = x1.y - qq.y;   // odd points
    accA = __builtin_amdgcn_wmma_f32_16x16x4_f32(false, Aa, false, Aa,
                                                 (short)0, accA, false, false);
    accB = __builtin_amdgcn_wmma_f32_16x16x4_f32(false, Ab, false, Ab,
                                                 (short)0, accB, false, false);
    x0 = nx0; x1 = nx1; qq = nqq;
  }
  {  // epilogue: final chunk (c0 = 252), compute only
    v2f Aa, Ab;
    Aa.x = x0.x - qq.x;  Aa.y = x1.x - qq.y;
    Ab.x = x0.y - qq.x;  Ab.y = x1.y - qq.y;
    accA = __builtin_amdgcn_wmma_f32_16x16x4_f32(false, Aa, false, Aa,
                                                 (short)0, accA, false, false);
    accB = __builtin_amdgcn_wmma_f32_16x16x4_f32(false, Ab, false, Ab,
                                                 (short)0, accB, false, false);
  }

  const float d2A = diagExtract(accA, lane);
  const float d2B = diagExtract(accB, lane);

  float bestV = -1.0f;
  int   bestI = 0x7FFFFFFF;
  if (lane < 16) {
    const int n = p0 + 2 * m;                 // chain A -> n, chain B -> n+1
    const float dA = sqrtf(d2A);
    const float dB = sqrtf(d2B);
    v2f old = *reinterpret_cast<const v2f*>(min_d + n);
    const float nmA = fminf(old.x, dA);
    const float nmB = fminf(old.y, dB);
    v2f nw; nw.x = nmA; nw.y = nmB;
    *reinterpret_cast<v2f*>(min_d + n) = nw;
    bestV = nmA; bestI = n;
    combineMax(bestV, bestI, nmB, n + 1);
  }
  waveReduceMax(bestV, bestI);
  if (lane == 0) { rv[wv] = bestV; ri[wv] = bestI; }
  __syncthreads();
  if (wv == 0) {
    float v = (lane < 8) ? rv[lane] : -1.0f;
    int   i = (lane < 8) ? ri[lane] : 0x7FFFFFFF;
    waveReduceMax(v, i);
    if (lane == 0) { pval[blockIdx.x] = v; pidx[blockIdx.x] = i; }
  }

  // Last-block global argmax; counter self-resets to 0 for the next round.
  __threadfence();
  if (tid == 0) {
    int prev = atomicAdd(counter, 1);
    lastFlag = (prev == (int)gridDim.x - 1) ? 1 : 0;
  }
  __syncthreads();
  if (lastFlag) {
    float v = -1.0f; int i = 0x7FFFFFFF;
    for (int j = tid; j < DIST_BLOCKS; j += 256) combineMax(v, i, pval[j], pidx[j]);
    waveReduceMax(v, i);
    if (lane == 0) { rv[wv] = v; ri[wv] = i; }
    __syncthreads();
    if (wv == 0) {
      float vv = (lane < 8) ? rv[lane] : -1.0f;
      int   ii = (lane < 8) ? ri[lane] : 0x7FFFFFFF;
      waveReduceMax(vv, ii);
      if (lane == 0) { indices[iter + 1] = ii; *counter = 0; }
    }
  }
}

// Final gathers: sampled feats [491,256], weights [491], indices as float [491].
__global__ void __launch_bounds__(256) coreset_gather(
    const float* __restrict__ F, const float* __restrict__ W,
    const int* __restrict__ indices, float* __restrict__ out) {
  const int s = blockIdx.x;    // 0..490
  const int c = threadIdx.x;   // 0..255
  const int n = indices[s];
  const int b = n >> 10, p = n & (HWSZ - 1);
  out[(size_t)s * CDIM + c] = F[(size_t)b * PLANE + (size_t)c * HWSZ + p];
  if (c == 0) {
    out[(size_t)NSAMP * CDIM + s]         = W[n];
    out[(size_t)NSAMP * CDIM + NSAMP + s] = (float)n;  // < 2^24, exact
  }
}

extern "C" void kernel_launch(void* const* d_in, const int* in_sizes, int n_in,
                              void* d_out, int out_size, void* d_ws, size_t ws_size,
                              hipStream_t stream) {
  const float* F = (const float*)d_in[0];   // features [96,256,32,32]
  const float* W = (const float*)d_in[1];   // weights  [98304]
  // d_in[2] = num_samples (fixed 491 by the reference constants)
  float* out = (float*)d_out;

  char* ws = (char*)d_ws;
  float* min_d   = (float*)(ws);                  // 98304 * 4 = 393216 B
  float* pval    = (float*)(ws + 393216);         // 384 * 4
  int*   pidx    = (int*)  (ws + 396288);         // 384 * 4
  int*   indices = (int*)  (ws + 399360);         // 512 * 4
  int*   counter = (int*)  (ws + 401408);         // 4

  coreset_init<<<NPTS / 256, 256, 0, stream>>>(min_d, indices, counter);
  for (int it = 0; it < NITER; ++it) {
    coreset_step<<<DIST_BLOCKS, 256, 0, stream>>>(F, min_d, pval, pidx,
                                                  indices, counter, it);
  }
  coreset_gather<<<NSAMP, 256, 0, stream>>>(F, W, indices, out);
}